// Encoder_22033182228818
// MI455X (gfx1250) — compile-verified
//
#include <hip/hip_runtime.h>
#include <math.h>

#define D_FEAT 128
#define HID 16

typedef float v2f __attribute__((ext_vector_type(2)));
typedef float v8f __attribute__((ext_vector_type(8)));

__device__ __forceinline__ v8f wmma_f32_k4(v2f a, v2f b, v8f c) {
  // D = A(16x4 f32) * B(4x16 f32) + C(16x16 f32)
  return __builtin_amdgcn_wmma_f32_16x16x4_f32(
      /*neg_a=*/false, a, /*neg_b=*/false, b,
      /*c_mod=*/(short)0, c, /*reuse_a=*/false, /*reuse_b=*/false);
}

// ---------------------------------------------------------------- zero fill
__global__ __launch_bounds__(256) void zero_f4(float4* __restrict__ p, int n4) {
  int i = blockIdx.x * blockDim.x + threadIdx.x;
  int stride = gridDim.x * blockDim.x;
  float4 z = make_float4(0.f, 0.f, 0.f, 0.f);
  for (; i < n4; i += stride) p[i] = z;
}

// ------------------------------------------------- edge scatter, d = 128
// one wave per edge; lane l handles features [4l, 4l+4)
__global__ __launch_bounds__(256) void scatter_add_128(
    const float* __restrict__ x, const int* __restrict__ src,
    const int* __restrict__ dst, float* __restrict__ agg, int n_edges) {
  long long t = (long long)blockIdx.x * blockDim.x + threadIdx.x;
  int e = (int)(t >> 5);
  if (e >= n_edges) return;
  int lane = (int)(t & 31);
  int s = src[e], d = dst[e];
  float4 v = ((const float4*)(x + (long long)s * D_FEAT))[lane];
  float* ag = agg + (long long)d * D_FEAT + lane * 4;
  atomicAdd(ag + 0, v.x);
  atomicAdd(ag + 1, v.y);
  atomicAdd(ag + 2, v.z);
  atomicAdd(ag + 3, v.w);
}

// ------------------------------------------------- edge scatter, d = 16
// 4 threads per edge; thread c handles features [4c, 4c+4)
__global__ __launch_bounds__(256) void scatter_add_16(
    const float* __restrict__ h, const int* __restrict__ src,
    const int* __restrict__ dst, float* __restrict__ agg, int n_edges) {
  long long t = (long long)blockIdx.x * blockDim.x + threadIdx.x;
  int e = (int)(t >> 2);
  if (e >= n_edges) return;
  int c = (int)(t & 3);
  int s = src[e], d = dst[e];
  float4 v = ((const float4*)(h + (long long)s * HID))[c];
  float* ag = agg + (long long)d * HID + c * 4;
  atomicAdd(ag + 0, v.x);
  atomicAdd(ag + 1, v.y);
  atomicAdd(ag + 2, v.z);
  atomicAdd(ag + 3, v.w);
}

// ------------------------------------------------- layer-0 GIN MLP (128->16->16)
// one wave32 per 16-node tile; full-precision WMMA 16x16x4 accumulation
__global__ __launch_bounds__(32) void gin_mlp_l0(
    const float* __restrict__ x, const float* __restrict__ agg,
    const float* __restrict__ W1, const float* __restrict__ B1,
    const float* __restrict__ W2, const float* __restrict__ B2,
    float* __restrict__ hout, int n_nodes) {
  __shared__ float zs[16][132];   // padded: conflict-free ds_load_b64 A-frags
  __shared__ float zs2[16][20];
  int lane = threadIdx.x;
  int node0 = blockIdx.x * 16;

  // stage z = x + agg for 16 nodes x 128 feats (coalesced float4)
  for (int idx = lane; idx < 16 * 32; idx += 32) {
    int row = idx >> 5, c4 = idx & 31;
    int node = node0 + row; if (node >= n_nodes) node = n_nodes - 1;
    float4 xv = ((const float4*)(x + (long long)node * D_FEAT))[c4];
    float4 av = ((const float4*)(agg + (long long)node * D_FEAT))[c4];
    float* zp = &zs[row][c4 * 4];
    zp[0] = xv.x + av.x; zp[1] = xv.y + av.y;
    zp[2] = xv.z + av.z; zp[3] = xv.w + av.w;
  }
  __syncthreads();

  int n = lane & 15;       // A: row M; B/C: col N
  int hi = lane >> 4;      // half-wave selects K pair / upper M rows
  v8f c = {0.f, 0.f, 0.f, 0.f, 0.f, 0.f, 0.f, 0.f};
#pragma unroll
  for (int s = 0; s < 32; ++s) {     // K = 128 in steps of 4
    int k = 4 * s + 2 * hi;
    v2f a; a.x = zs[n][k]; a.y = zs[n][k + 1];
    v2f b; b.x = W1[k * HID + n]; b.y = W1[(k + 1) * HID + n];
    c = wmma_f32_k4(a, b, c);
  }
  float b1v = B1[n];
#pragma unroll
  for (int r = 0; r < 8; ++r) {      // bias + ReLU, transpose via LDS
    zs2[r + 8 * hi][n] = fmaxf(c[r] + b1v, 0.f);
  }
  __syncthreads();

  v8f c2 = {0.f, 0.f, 0.f, 0.f, 0.f, 0.f, 0.f, 0.f};
#pragma unroll
  for (int s = 0; s < 4; ++s) {      // K = 16
    int k = 4 * s + 2 * hi;
    v2f a; a.x = zs2[n][k]; a.y = zs2[n][k + 1];
    v2f b; b.x = W2[k * HID + n]; b.y = W2[(k + 1) * HID + n];
    c2 = wmma_f32_k4(a, b, c2);
  }
  float b2v = B2[n];
#pragma unroll
  for (int r = 0; r < 8; ++r) {
    int node = node0 + r + 8 * hi;
    if (node < n_nodes)              // inter-layer ReLU after conv 0
      hout[(long long)node * HID + n] = fmaxf(c2[r] + b2v, 0.f);
  }
}

// ------------------------------------------------- hidden GIN MLP (16->16->16)
__global__ __launch_bounds__(32) void gin_mlp_h(
    const float* __restrict__ h, const float* __restrict__ agg,
    const float* __restrict__ W1, const float* __restrict__ B1,
    const float* __restrict__ W2, const float* __restrict__ B2,
    float* __restrict__ hout, int n_nodes, int relu_out) {
  __shared__ float zs[16][20];
  __shared__ float zs2[16][20];
  int lane = threadIdx.x;
  int node0 = blockIdx.x * 16;

  for (int idx = lane; idx < 16 * 4; idx += 32) {
    int row = idx >> 2, c4 = idx & 3;
    int node = node0 + row; if (node >= n_nodes) node = n_nodes - 1;
    float4 hv = ((const float4*)(h + (long long)node * HID))[c4];
    float4 av = ((const float4*)(agg + (long long)node * HID))[c4];
    float* zp = &zs[row][c4 * 4];
    zp[0] = hv.x + av.x; zp[1] = hv.y + av.y;
    zp[2] = hv.z + av.z; zp[3] = hv.w + av.w;
  }
  __syncthreads();

  int n = lane & 15;
  int hi = lane >> 4;
  v8f c = {0.f, 0.f, 0.f, 0.f, 0.f, 0.f, 0.f, 0.f};
#pragma unroll
  for (int s = 0; s < 4; ++s) {
    int k = 4 * s + 2 * hi;
    v2f a; a.x = zs[n][k]; a.y = zs[n][k + 1];
    v2f b; b.x = W1[k * HID + n]; b.y = W1[(k + 1) * HID + n];
    c = wmma_f32_k4(a, b, c);
  }
  float b1v = B1[n];
#pragma unroll
  for (int r = 0; r < 8; ++r) zs2[r + 8 * hi][n] = fmaxf(c[r] + b1v, 0.f);
  __syncthreads();

  v8f c2 = {0.f, 0.f, 0.f, 0.f, 0.f, 0.f, 0.f, 0.f};
#pragma unroll
  for (int s = 0; s < 4; ++s) {
    int k = 4 * s + 2 * hi;
    v2f a; a.x = zs2[n][k]; a.y = zs2[n][k + 1];
    v2f b; b.x = W2[k * HID + n]; b.y = W2[(k + 1) * HID + n];
    c2 = wmma_f32_k4(a, b, c2);
  }
  float b2v = B2[n];
#pragma unroll
  for (int r = 0; r < 8; ++r) {
    int node = node0 + r + 8 * hi;
    if (node < n_nodes) {
      float v = c2[r] + b2v;
      if (relu_out) v = fmaxf(v, 0.f);
      hout[(long long)node * HID + n] = v;
    }
  }
}

// ------------------------------------------------- head: mean / softplus(std)
__global__ __launch_bounds__(256) void head_kernel(
    const float* __restrict__ h, const float* __restrict__ mw,
    const float* __restrict__ mb, const float* __restrict__ sw,
    const float* __restrict__ sb, float* __restrict__ out, int n_nodes) {
  int i = blockIdx.x * blockDim.x + threadIdx.x;
  if (i >= n_nodes) return;
  float hv[16];
#pragma unroll
  for (int k = 0; k < 16; ++k) hv[k] = h[(long long)i * HID + k];
#pragma unroll
  for (int j = 0; j < 8; ++j) {
    float m = mb[j], s = sb[j];
#pragma unroll
    for (int k = 0; k < 16; ++k) {
      m = fmaf(hv[k], mw[k * 8 + j], m);
      s = fmaf(hv[k], sw[k * 8 + j], s);
    }
    out[(long long)i * 8 + j] = m;
    // stable softplus
    out[(long long)n_nodes * 8 + (long long)i * 8 + j] =
        fmaxf(s, 0.f) + log1pf(expf(-fabsf(s)));
  }
}

extern "C" void kernel_launch(void* const* d_in, const int* in_sizes, int n_in,
                              void* d_out, int out_size, void* d_ws, size_t ws_size,
                              hipStream_t stream) {
  const float* x    = (const float*)d_in[0];
  const int*   ei   = (const int*)d_in[1];
  const float* w1_0 = (const float*)d_in[2];
  const float* b1_0 = (const float*)d_in[3];
  const float* w2_0 = (const float*)d_in[4];
  const float* b2_0 = (const float*)d_in[5];
  const float* w1   = (const float*)d_in[6];
  const float* b1   = (const float*)d_in[7];
  const float* w2   = (const float*)d_in[8];
  const float* b2   = (const float*)d_in[9];
  const float* mw   = (const float*)d_in[10];
  const float* mb   = (const float*)d_in[11];
  const float* sw   = (const float*)d_in[12];
  const float* sb   = (const float*)d_in[13];

  int n_nodes = in_sizes[0] / D_FEAT;
  int n_edges = in_sizes[1] / 2;
  const int* src = ei;
  const int* dst = ei + n_edges;

  char* ws = (char*)d_ws;
  float* agg128 = (float*)ws; ws += (size_t)n_nodes * D_FEAT * sizeof(float);
  float* agg16  = (float*)ws; ws += (size_t)n_nodes * HID * sizeof(float);
  float* hA     = (float*)ws; ws += (size_t)n_nodes * HID * sizeof(float);
  float* hB     = (float*)ws;

  int tiles = (n_nodes + 15) / 16;

  // ---- layer 0 (d = 128)
  zero_f4<<<2048, 256, 0, stream>>>((float4*)agg128, n_nodes * D_FEAT / 4);
  {
    long long tt = (long long)n_edges * 32;
    int blocks = (int)((tt + 255) / 256);
    scatter_add_128<<<blocks, 256, 0, stream>>>(x, src, dst, agg128, n_edges);
  }
  gin_mlp_l0<<<tiles, 32, 0, stream>>>(x, agg128, w1_0, b1_0, w2_0, b2_0, hA, n_nodes);

  // ---- layers 1..9 (d = 16), ping-pong
  float* hin = hA;
  float* hout = hB;
  for (int i = 0; i < 9; ++i) {
    zero_f4<<<1024, 256, 0, stream>>>((float4*)agg16, n_nodes * HID / 4);
    long long tt = (long long)n_edges * 4;
    int blocks = (int)((tt + 255) / 256);
    scatter_add_16<<<blocks, 256, 0, stream>>>(hin, src, dst, agg16, n_edges);
    int relu_out = (i < 8) ? 1 : 0;
    gin_mlp_h<<<tiles, 32, 0, stream>>>(hin, agg16, w1 + i * 256, b1 + i * 16,
                                        w2 + i * 256, b2 + i * 16, hout,
                                        n_nodes, relu_out);
    float* t = hin; hin = hout; hout = t;
  }

  // ---- head
  head_kernel<<<(n_nodes + 255) / 256, 256, 0, stream>>>(
      hin, mw, mb, sw, sb, (float*)d_out, n_nodes);
}